// B_Conv2d_ConvNN_Spatial_K_N_20435454394604
// MI455X (gfx1250) — compile-verified
//
#include <hip/hip_runtime.h>

typedef __attribute__((ext_vector_type(2))) float v2f;
typedef __attribute__((ext_vector_type(8))) float v8f;

#define BATCH 256

// ---------------------------------------------------------------------------
// helper: read element of pixel_unshuffle(src, 2):
// u[b][cc][y][x] = src[b][cc>>2][2y + ((cc>>1)&1)][2x + (cc&1)], 16x16 spatial
// p = y*16 + x
// ---------------------------------------------------------------------------
__device__ __forceinline__ float load_unshuf(const float* __restrict__ src,
                                             int b, int Csrc, int cc, int p) {
  int y = p >> 4, x = p & 15;
  int c = cc >> 2, i = (cc >> 1) & 1, j = cc & 1;
  return src[((b * Csrc + c) * 32 + (y * 2 + i)) * 32 + (x * 2 + j)];
}

// flat byte offset (within src image tensor) of unshuffled element
__device__ __forceinline__ unsigned unshuf_goff(int b, int Csrc, int cc, int p) {
  int y = p >> 4, x = p & 15;
  int c = cc >> 2, i = (cc >> 1) & 1, j = cc & 1;
  return (unsigned)((((b * Csrc + c) * 32 + (y * 2 + i)) * 32 + (x * 2 + j)) * 4);
}

// LDS byte offset of a __shared__ object (generic LDS pointer low 32 bits)
__device__ __forceinline__ unsigned lds_off_of(const void* p) {
  return (unsigned)(size_t)p;
}

// async gather: copies 4 bytes from the global address base + gOff to the LDS offset ldsOff, tracked by ASYNCcnt
__device__ __forceinline__ void async_gather_b32(unsigned ldsOff, unsigned gOff,
                                                 const void* base) {
  asm volatile("global_load_async_to_lds_b32 %0, %1, %2"
               :: "v"(ldsOff), "v"(gOff), "s"(base)
               : "memory");
}

__device__ __forceinline__ void wait_asynccnt0() {
  asm volatile("s_wait_asynccnt 0x0" ::: "memory");
}

// ---------------------------------------------------------------------------
// 3x3 SAME conv, direct. Writes (with bias) into cat buffer channels [0,Cout).
// ---------------------------------------------------------------------------
__global__ void conv3x3_kernel(const float* __restrict__ src,
                               const float* __restrict__ w,
                               const float* __restrict__ bias,
                               float* __restrict__ cat,
                               int Cin, int Cout, int Ccat) {
  int idx = blockIdx.x * 256 + threadIdx.x;
  int p = idx & 1023;
  int t = idx >> 10;
  int co = t % Cout;
  int b  = t / Cout;
  if (b >= BATCH) return;
  int y = p >> 5, x = p & 31;
  float s = bias[co];
  for (int ci = 0; ci < Cin; ++ci) {
    const float* sp = src + ((b * Cin + ci) * 32) * 32;
    const float* wp = w + ((co * Cin + ci) * 3) * 3;
#pragma unroll
    for (int dy = 0; dy < 3; ++dy) {
      int yy = y + dy - 1;
      if ((unsigned)yy >= 32u) continue;
#pragma unroll
      for (int dx = 0; dx < 3; ++dx) {
        int xx = x + dx - 1;
        if ((unsigned)xx >= 32u) continue;
        s += sp[yy * 32 + xx] * wp[dy * 3 + dx];
      }
    }
  }
  cat[(b * Ccat + co) * 1024 + p] = s;
}

// ---------------------------------------------------------------------------
// Per-pixel neighbor ordering for _convnn_spatial.
// One block per image, one thread per pixel (256). Computes distances to the
// 8 sampled rows, stable-sorts ascending (== lax.top_k(-d)), writes a 9-entry
// neighbor pixel table: [p, smp[ord0], ..., smp[ord7]].
// ---------------------------------------------------------------------------
__global__ void nn_order_kernel(const float* __restrict__ src, int Csrc,
                                int* __restrict__ nbr) {
  const int smp[8] = {0, 36, 72, 109, 145, 182, 218, 255};
  int b = blockIdx.x;
  int p = threadIdx.x;
  int CC = Csrc * 4;
  __shared__ float s_lds[8 * 64];  // sample rows, stride 64
  for (int i = threadIdx.x; i < 8 * CC; i += 256) {
    int m = i / CC, c = i % CC;
    s_lds[m * 64 + c] = load_unshuf(src, b, Csrc, c, smp[m]);
  }
  __syncthreads();
  float d[8];
#pragma unroll
  for (int m = 0; m < 8; ++m) d[m] = 0.f;
  for (int c = 0; c < CC; ++c) {
    float tc = load_unshuf(src, b, Csrc, c, p);
#pragma unroll
    for (int m = 0; m < 8; ++m) {
      float df = tc - s_lds[m * 64 + c];
      d[m] += df * df;
    }
  }
  // stable insertion sort ascending by distance (ties -> lower index first)
  float dv[8];
  int ov[8];
#pragma unroll
  for (int m = 0; m < 8; ++m) { dv[m] = d[m]; ov[m] = m; }
#pragma unroll
  for (int m = 1; m < 8; ++m) {
    float kd = dv[m];
    int ko = ov[m];
    int q = m - 1;
    while (q >= 0 && dv[q] > kd) { dv[q + 1] = dv[q]; ov[q + 1] = ov[q]; --q; }
    dv[q + 1] = kd; ov[q + 1] = ko;
  }
  int* np = nbr + (b * 256 + p) * 9;
  np[0] = p;
#pragma unroll
  for (int r = 0; r < 8; ++r) np[1 + r] = smp[ov[r]];
}

// ---------------------------------------------------------------------------
// nn weight transpose: Bt[j*Nn + o] = w[o][c][k], j = k*CC + c
// ---------------------------------------------------------------------------
__global__ void transpose_nnw_kernel(const float* __restrict__ w,
                                     float* __restrict__ Bt, int CC, int Nn) {
  int idx = blockIdx.x * 256 + threadIdx.x;
  int total = 9 * CC * Nn;
  if (idx >= total) return;
  int o = idx % Nn;
  int j = idx / Nn;
  int k = j / CC, c = j % CC;
  Bt[j * Nn + o] = w[(o * CC + c) * 9 + k];
}

// ---------------------------------------------------------------------------
// _convnn_spatial main GEMM via V_WMMA_F32_16X16X4_F32.
// One block per image (8 waves). t (unshuffled image, CC x 256) and the
// neighbor table are gathered straight into LDS with
// GLOBAL_LOAD_ASYNC_TO_LDS_B32 (ASYNCcnt path, no VGPR round trip).
// A rows are gathered per-pixel via the neighbor table; B is the
// pre-transposed weight matrix. Output is written with fused pixel_shuffle +
// bias into the cat buffer at channel offset coff.
// ---------------------------------------------------------------------------
__global__ __launch_bounds__(256) void convnn_wmma_kernel(
    const float* __restrict__ src, int Csrc, const int* __restrict__ nbr,
    const float* __restrict__ Bt, const float* __restrict__ nnb,
    float* __restrict__ cat, int Ccat, int coff, int Nn) {
  int b = blockIdx.x;
  int CC = Csrc * 4;
  __shared__ float t_lds[64 * 256];
  __shared__ int nbr_lds[256 * 9];

  // async gather of the unshuffled image into LDS
  unsigned tBase = lds_off_of(&t_lds[0]);
  for (int i = threadIdx.x; i < CC * 256; i += 256) {
    int cc = i >> 8, p = i & 255;
    async_gather_b32(tBase + (unsigned)(i * 4), unshuf_goff(b, Csrc, cc, p), src);
  }
  // async copy of the neighbor table into LDS
  unsigned nBase = lds_off_of(&nbr_lds[0]);
  for (int i = threadIdx.x; i < 256 * 9; i += 256) {
    async_gather_b32(nBase + (unsigned)(i * 4), (unsigned)((b * 2304 + i) * 4), nbr);
  }
  wait_asynccnt0();
  __syncthreads();

  int wave = threadIdx.x >> 5;
  int lane = threadIdx.x & 31;
  int half = lane >> 4;      // K sub-fragment select
  int l    = lane & 15;      // row (A) / column (B,D) index
  int ntiles = (Nn >> 4) * 16;
  for (int tile = wave; tile < ntiles; tile += 8) {
    int m0 = (tile & 15) * 16;  // pixel tile
    int n0 = (tile >> 4) * 16;  // out-channel tile
    int pixel = m0 + l;
    int col = n0 + l;
    v8f acc0 = {}, acc1 = {};
    for (int k = 0; k < 9; ++k) {
      int np = nbr_lds[pixel * 9 + k];
      const float* tp = t_lds + np;
      int c0 = 0;
      // dual-accumulator main body: 4 independent loads before first WMMA
      for (; c0 + 8 <= CC; c0 += 8) {
        int ca = c0 + 2 * half;
        int cb = c0 + 4 + 2 * half;
        v2f a0, a1, b0, b1;
        a0.x = tp[ca * 256];       a0.y = tp[(ca + 1) * 256];
        a1.x = tp[cb * 256];       a1.y = tp[(cb + 1) * 256];
        int ja = k * CC + ca, jb = k * CC + cb;
        b0.x = Bt[ja * Nn + col];  b0.y = Bt[(ja + 1) * Nn + col];
        b1.x = Bt[jb * Nn + col];  b1.y = Bt[(jb + 1) * Nn + col];
        acc0 = __builtin_amdgcn_wmma_f32_16x16x4_f32(false, a0, false, b0,
                                                     (short)0, acc0, false, false);
        acc1 = __builtin_amdgcn_wmma_f32_16x16x4_f32(false, a1, false, b1,
                                                     (short)0, acc1, false, false);
      }
      // remainder (CC % 8 == 4 case, e.g. CC = 12)
      for (; c0 < CC; c0 += 4) {
        int ca = c0 + 2 * half;
        v2f a0, b0;
        a0.x = tp[ca * 256];       a0.y = tp[(ca + 1) * 256];
        int ja = k * CC + ca;
        b0.x = Bt[ja * Nn + col];  b0.y = Bt[(ja + 1) * Nn + col];
        acc0 = __builtin_amdgcn_wmma_f32_16x16x4_f32(false, a0, false, b0,
                                                     (short)0, acc0, false, false);
      }
    }
    v8f acc = acc0 + acc1;
    // D[r] holds out[pixel = m0 + r + 8*half][channel = n0 + l]
#pragma unroll
    for (int r = 0; r < 8; ++r) {
      int o = n0 + l;
      int p = m0 + r + 8 * half;
      float v = acc[r] + nnb[o];
      int ct = o >> 2, oi = (o >> 1) & 1, oj = o & 1;
      int py = p >> 4, px = p & 15;
      cat[((b * Ccat + coff + ct) * 32 + (py * 2 + oi)) * 32 + (px * 2 + oj)] = v;
    }
  }
}

// ---------------------------------------------------------------------------
// pointwise 1x1 conv + ReLU over cat buffer
// ---------------------------------------------------------------------------
__global__ void pointwise_relu_kernel(const float* __restrict__ cat,
                                      const float* __restrict__ pw,
                                      const float* __restrict__ pb,
                                      float* __restrict__ dst, int Ccat, int O) {
  int idx = blockIdx.x * 256 + threadIdx.x;
  int p = idx & 1023;
  int t = idx >> 10;
  int o = t % O;
  int b = t / O;
  if (b >= BATCH) return;
  float s = pb[o];
  for (int c = 0; c < Ccat; ++c)
    s += cat[(b * Ccat + c) * 1024 + p] * pw[o * Ccat + c];
  dst[(b * O + o) * 1024 + p] = fmaxf(s, 0.f);
}

// ---------------------------------------------------------------------------
// fc1: out[b][m] = relu(sum_k h2[b][k] * W[m][k] + bias[m])
// M=1024 out channels, N=256 batch, K=32768. One wave per 16x16 tile.
// Both operands stream K contiguously per lane (cacheline efficient).
// 4 independent accumulators per K-step-16 + prefetch 8KB ahead to keep the
// HBM stream saturated (this GEMM is bandwidth-bound: 134MB of fp32 weights).
// ---------------------------------------------------------------------------
__global__ __launch_bounds__(32) void fc1_wmma_kernel(
    const float* __restrict__ h2, const float* __restrict__ w,
    const float* __restrict__ bias, float* __restrict__ out) {
  int m0 = blockIdx.x * 16;   // out-channel tile
  int n0 = blockIdx.y * 16;   // batch tile
  int lane = threadIdx.x;
  int half = lane >> 4;
  int l = lane & 15;
  const float* arow = w  + (size_t)(m0 + l) * 32768 + 2 * half;
  const float* brow = h2 + (size_t)(n0 + l) * 32768 + 2 * half;
  v8f acc0 = {}, acc1 = {}, acc2 = {}, acc3 = {};
  for (int k0 = 0; k0 < 32768; k0 += 16) {
    v2f a0 = *(const v2f*)(arow + k0);
    v2f b0 = *(const v2f*)(brow + k0);
    v2f a1 = *(const v2f*)(arow + k0 + 4);
    v2f b1 = *(const v2f*)(brow + k0 + 4);
    v2f a2 = *(const v2f*)(arow + k0 + 8);
    v2f b2 = *(const v2f*)(brow + k0 + 8);
    v2f a3 = *(const v2f*)(arow + k0 + 12);
    v2f b3 = *(const v2f*)(brow + k0 + 12);
    __builtin_prefetch(arow + k0 + 2048, 0, 0);
    __builtin_prefetch(brow + k0 + 2048, 0, 0);
    acc0 = __builtin_amdgcn_wmma_f32_16x16x4_f32(false, a0, false, b0,
                                                 (short)0, acc0, false, false);
    acc1 = __builtin_amdgcn_wmma_f32_16x16x4_f32(false, a1, false, b1,
                                                 (short)0, acc1, false, false);
    acc2 = __builtin_amdgcn_wmma_f32_16x16x4_f32(false, a2, false, b2,
                                                 (short)0, acc2, false, false);
    acc3 = __builtin_amdgcn_wmma_f32_16x16x4_f32(false, a3, false, b3,
                                                 (short)0, acc3, false, false);
  }
  v8f acc = (acc0 + acc1) + (acc2 + acc3);
#pragma unroll
  for (int r = 0; r < 8; ++r) {
    int m = m0 + r + 8 * half;
    float v = acc[r] + bias[m];
    out[(n0 + l) * 1024 + m] = fmaxf(v, 0.f);
  }
}

// ---------------------------------------------------------------------------
// fc2: (256,1024) x (10,1024)^T + bias -> (256,10)
// ---------------------------------------------------------------------------
__global__ void fc2_kernel(const float* __restrict__ fc1o,
                           const float* __restrict__ w,
                           const float* __restrict__ bias,
                           float* __restrict__ out) {
  int b = threadIdx.x;
  int o = blockIdx.x;
  const float* hrow = fc1o + b * 1024;
  const float* wrow = w + o * 1024;
  float s = bias[o];
#pragma unroll 4
  for (int k = 0; k < 1024; ++k) s += hrow[k] * wrow[k];
  out[b * 10 + o] = s;
}

// ---------------------------------------------------------------------------
// launch
// ---------------------------------------------------------------------------
extern "C" void kernel_launch(void* const* d_in, const int* in_sizes, int n_in,
                              void* d_out, int out_size, void* d_ws, size_t ws_size,
                              hipStream_t stream) {
  (void)in_sizes; (void)n_in; (void)out_size; (void)ws_size;
  const float* x       = (const float*)d_in[0];
  const float* conv1_w = (const float*)d_in[1];
  const float* conv1_b = (const float*)d_in[2];
  const float* nn1_w   = (const float*)d_in[3];
  const float* nn1_b   = (const float*)d_in[4];
  const float* pw1_w   = (const float*)d_in[5];
  const float* pw1_b   = (const float*)d_in[6];
  const float* conv2_w = (const float*)d_in[7];
  const float* conv2_b = (const float*)d_in[8];
  const float* nn2_w   = (const float*)d_in[9];
  const float* nn2_b   = (const float*)d_in[10];
  const float* pw2_w   = (const float*)d_in[11];
  const float* pw2_b   = (const float*)d_in[12];
  const float* fc1_w   = (const float*)d_in[13];
  const float* fc1_b   = (const float*)d_in[14];
  const float* fc2_w   = (const float*)d_in[15];
  const float* fc2_b   = (const float*)d_in[16];

  char* ws = (char*)d_ws;
  // workspace layout (bytes)
  const size_t OFF_H1   = 0;                         // 256*16*1024 f32 = 16 MiB
  const size_t OFF_H2   = OFF_H1 + 16777216ull;      // 256*32*1024 f32 = 32 MiB
  const size_t OFF_CAT  = OFF_H2 + 33554432ull;      // 256*64*1024 f32 = 64 MiB
  const size_t OFF_FC1O = OFF_CAT + 67108864ull;     // 256*1024  f32 =  1 MiB
  const size_t OFF_NBR  = OFF_FC1O + 1048576ull;     // 256*256*9 i32
  const size_t OFF_BT   = OFF_NBR + 2359296ull;      // 576*128   f32
  float* h1   = (float*)(ws + OFF_H1);
  float* h2   = (float*)(ws + OFF_H2);
  float* cat  = (float*)(ws + OFF_CAT);
  float* fc1o = (float*)(ws + OFF_FC1O);
  int*   nbr  = (int*)(ws + OFF_NBR);
  float* Bt   = (float*)(ws + OFF_BT);

  // ---- branch 1 (x: 256x3x32x32 -> h1: 256x16x32x32) ----
  conv3x3_kernel<<<BATCH * 16 * 1024 / 256, 256, 0, stream>>>(
      x, conv1_w, conv1_b, cat, 3, 16, 32);
  nn_order_kernel<<<BATCH, 256, 0, stream>>>(x, 3, nbr);
  transpose_nnw_kernel<<<(9 * 12 * 64 + 255) / 256, 256, 0, stream>>>(
      nn1_w, Bt, 12, 64);
  convnn_wmma_kernel<<<BATCH, 256, 0, stream>>>(
      x, 3, nbr, Bt, nn1_b, cat, 32, 16, 64);
  pointwise_relu_kernel<<<BATCH * 16 * 1024 / 256, 256, 0, stream>>>(
      cat, pw1_w, pw1_b, h1, 32, 16);

  // ---- branch 2 (h1 -> h2: 256x32x32x32) ----
  conv3x3_kernel<<<BATCH * 32 * 1024 / 256, 256, 0, stream>>>(
      h1, conv2_w, conv2_b, cat, 16, 32, 64);
  nn_order_kernel<<<BATCH, 256, 0, stream>>>(h1, 16, nbr);
  transpose_nnw_kernel<<<(9 * 64 * 128 + 255) / 256, 256, 0, stream>>>(
      nn2_w, Bt, 64, 128);
  convnn_wmma_kernel<<<BATCH, 256, 0, stream>>>(
      h1, 16, nbr, Bt, nn2_b, cat, 64, 32, 128);
  pointwise_relu_kernel<<<BATCH * 32 * 1024 / 256, 256, 0, stream>>>(
      cat, pw2_w, pw2_b, h2, 64, 32);

  // ---- fc1 (memory-bound 256x32768x1024 GEMM, fp32 WMMA) + ReLU ----
  fc1_wmma_kernel<<<dim3(64, 16), 32, 0, stream>>>(h2, fc1_w, fc1_b, fc1o);

  // ---- fc2 ----
  fc2_kernel<<<10, 256, 0, stream>>>(fc1o, fc2_w, fc2_b, (float*)d_out);
}